// ProgressiveL3Layer_23785528885991
// MI455X (gfx1250) — compile-verified
//
#include <hip/hip_runtime.h>

typedef __attribute__((ext_vector_type(2))) float v2f;
typedef __attribute__((ext_vector_type(8))) float v8f;

constexpr int D_MODEL = 1024;
constexpr int D_FF    = 4096;
constexpr int N_EMB   = 16384;
constexpr int NTOK    = 2048;      // B*T
constexpr float LN_EPS = 1e-5f;

// workspace layout (float offsets)
constexpr size_t OFF_SLOTS  = 0;        // 64 floats: 0=x amax,1=WK,2=WV,3=Wskip,4=hidden
constexpr size_t OFF_COLMAX = 64;       // 4096 (W_in per-column amax)
constexpr size_t OFF_ROWMAX = 64 + 4096;            // 4096 (W_out per-row amax)
constexpr size_t OFF_MEAN   = OFF_ROWMAX + 4096;    // 2048
constexpr size_t OFF_RSTD   = OFF_MEAN + 2048;      // 2048
constexpr size_t OFF_A      = 16384;                          // 2048*1024
constexpr size_t OFF_COMB   = OFF_A + (size_t)NTOK * D_MODEL; // 2048*1024
constexpr size_t OFF_HIDDEN = OFF_COMB + (size_t)NTOK * D_MODEL; // 2048*4096

__device__ __forceinline__ float wave_max(float v) {
  #pragma unroll
  for (int o = 16; o > 0; o >>= 1) v = fmaxf(v, __shfl_xor(v, o, 32));
  return v;
}
__device__ __forceinline__ float wave_sum(float v) {
  #pragma unroll
  for (int o = 16; o > 0; o >>= 1) v += __shfl_xor(v, o, 32);
  return v;
}

__global__ void init_slots_kernel(float* slots) {
  if (threadIdx.x < 64) slots[threadIdx.x] = 0.0f;
}

// grid-stride |.| max into slot via atomicMax on float bits (values >= 0)
__global__ __launch_bounds__(256) void amax_kernel(const float* __restrict__ p, long n,
                                                   float* __restrict__ slot) {
  __shared__ float red[8];
  float m = 0.0f;
  for (long i = (long)blockIdx.x * blockDim.x + threadIdx.x; i < n;
       i += (long)gridDim.x * blockDim.x)
    m = fmaxf(m, fabsf(p[i]));
  m = wave_max(m);
  const int wid = threadIdx.x >> 5, lane = threadIdx.x & 31;
  if (lane == 0) red[wid] = m;
  __syncthreads();
  if (threadIdx.x == 0) {
    float mm = red[0];
    #pragma unroll
    for (int i = 1; i < 8; ++i) mm = fmaxf(mm, red[i]);
    atomicMax((unsigned int*)slot, __float_as_uint(mm));
  }
}

// per-column abs-max of W_in [D_MODEL x D_FF] (axis=0); thread per column, coalesced
__global__ __launch_bounds__(256) void colmax_kernel(const float* __restrict__ W_in,
                                                     float* __restrict__ colmax) {
  const int f = blockIdx.x * 256 + threadIdx.x;
  float m = 0.0f;
  for (int d = 0; d < D_MODEL; ++d) m = fmaxf(m, fabsf(W_in[(size_t)d * D_FF + f]));
  colmax[f] = m;
}

// per-row abs-max of W_out [D_FF x D_MODEL] (axis=1); wave per row
__global__ __launch_bounds__(256) void rowmax_kernel(const float* __restrict__ W_out,
                                                     float* __restrict__ rowmax) {
  const int row = blockIdx.x * 8 + (threadIdx.x >> 5);
  const int lane = threadIdx.x & 31;
  const float* p = W_out + (size_t)row * D_MODEL;
  float m = 0.0f;
  for (int k = lane; k < D_MODEL; k += 32) m = fmaxf(m, fabsf(p[k]));
  m = wave_max(m);
  if (lane == 0) rowmax[row] = m;
}

// A = layernorm(quantize_a16(x)) * w + b ; block per token
__global__ __launch_bounds__(256) void ln_in_kernel(const float* __restrict__ x,
                                                    const float* __restrict__ slots,
                                                    const float* __restrict__ w,
                                                    const float* __restrict__ b,
                                                    float* __restrict__ A) {
  __shared__ float xs_s[D_MODEL];
  __shared__ float red[16];
  const int t = blockIdx.x;
  const float sx = 32767.0f / fmaxf(slots[0], 1e-8f), sxi = 1.0f / sx;
  float s1 = 0.f, s2 = 0.f;
  for (int d = threadIdx.x; d < D_MODEL; d += 256) {
    float xs = rintf(x[(size_t)t * D_MODEL + d] * sx) * sxi;
    xs_s[d] = xs; s1 += xs; s2 += xs * xs;
  }
  s1 = wave_sum(s1); s2 = wave_sum(s2);
  const int wid = threadIdx.x >> 5, lane = threadIdx.x & 31;
  if (lane == 0) { red[wid] = s1; red[8 + wid] = s2; }
  __syncthreads();
  if (threadIdx.x == 0) {
    float a = 0.f, c = 0.f;
    #pragma unroll
    for (int i = 0; i < 8; ++i) { a += red[i]; c += red[8 + i]; }
    red[0] = a; red[8] = c;
  }
  __syncthreads();
  const float mean = red[0] * (1.0f / D_MODEL);
  const float var  = red[8] * (1.0f / D_MODEL) - mean * mean;
  const float rs   = rsqrtf(var + LN_EPS);
  for (int d = threadIdx.x; d < D_MODEL; d += 256)
    A[(size_t)t * D_MODEL + d] = (xs_s[d] - mean) * rs * w[d] + b[d];
}

// score/softmax/comb over contiguous 64-column slab; block per token
__global__ __launch_bounds__(256) void attn_kernel(const float* __restrict__ A,
                                                   const float* __restrict__ WK,
                                                   const float* __restrict__ WV,
                                                   const int* __restrict__ ids,
                                                   const float* __restrict__ slots,
                                                   float* __restrict__ comb) {
  __shared__ float a_s[D_MODEL];
  __shared__ float part[64 * 5];
  __shared__ float p_s[64];
  __shared__ float red2[4];
  const int t = blockIdx.x;
  const int c0 = (ids[t] & 255) << 6;  // (id*64) % 16384, never wraps within 64
  const float sk = 127.0f / fmaxf(slots[1], 1e-8f), ski = 1.0f / sk;
  const float sv = 127.0f / fmaxf(slots[2], 1e-8f), svi = 1.0f / sv;
  for (int d = threadIdx.x; d < D_MODEL; d += 256)
    a_s[d] = A[(size_t)t * D_MODEL + d];
  __syncthreads();
  const int kk = threadIdx.x & 63, dd = threadIdx.x >> 6;
  {
    float acc = 0.f;
    const float* wp = WK + c0 + kk;
    for (int d = dd * 256; d < dd * 256 + 256; ++d)
      acc += a_s[d] * (rintf(wp[(size_t)d * N_EMB] * sk) * ski);
    part[kk * 5 + dd] = acc;
  }
  __syncthreads();
  if (threadIdx.x < 64)
    p_s[threadIdx.x] = part[threadIdx.x * 5] + part[threadIdx.x * 5 + 1] +
                       part[threadIdx.x * 5 + 2] + part[threadIdx.x * 5 + 3];
  __syncthreads();
  float v = (threadIdx.x < 64) ? p_s[threadIdx.x] : -3.4e38f;
  if (threadIdx.x < 64) {
    float m = wave_max(v);
    if ((threadIdx.x & 31) == 0) red2[threadIdx.x >> 5] = m;
  }
  __syncthreads();
  const float gmax = fmaxf(red2[0], red2[1]);
  float e = (threadIdx.x < 64) ? __expf(v - gmax) : 0.f;
  if (threadIdx.x < 64) {
    float ssum = wave_sum(e);
    if ((threadIdx.x & 31) == 0) red2[2 + (threadIdx.x >> 5)] = ssum;
  }
  __syncthreads();
  const float denom = red2[2] + red2[3];
  if (threadIdx.x < 64) p_s[threadIdx.x] = e / denom;
  __syncthreads();
  for (int d = threadIdx.x; d < D_MODEL; d += 256) {
    const float* wp = WV + (size_t)d * N_EMB + c0;
    float acc = 0.f;
    #pragma unroll 8
    for (int k = 0; k < 64; ++k)
      acc += p_s[k] * (rintf(wp[k] * sv) * svi);
    comb[(size_t)t * D_MODEL + d] = acc;
  }
}

// hidden[2048,4096] = comb[2048,1024] @ dequant_col(W_in)
// block tile 64(M) x 128(N); 8 waves, each wave: 4 C tiles (one B frag feeds 4 WMMAs)
__global__ __launch_bounds__(256) void gemm_hidden_kernel(const float* __restrict__ comb,
                                                          const float* __restrict__ W_in,
                                                          const float* __restrict__ colmax,
                                                          float* __restrict__ hidden) {
  constexpr int MTILE = 64;
  constexpr int KCH = 128;
  constexpr int LDA = KCH + 2;           // lane m -> bank 2m: conflict-free b64 reads
  __shared__ float a_s[MTILE * LDA];     // 33,280 B
  const int tid = threadIdx.x;
  const int lane = tid & 31, wid = tid >> 5;
  const int m0 = blockIdx.x * MTILE;
  const int n0 = blockIdx.y * 128 + wid * 16;
  const int half = lane >> 4, l16 = lane & 15;
  const int nn = n0 + l16;
  const int koff = half * 2;

  const float cm = colmax[nn];
  const float s = 127.0f / fmaxf(cm, 1e-8f);
  const float sinv = 1.0f / s;

  v8f acc[4];
  #pragma unroll
  for (int ms = 0; ms < 4; ++ms) acc[ms] = v8f{0.f, 0.f, 0.f, 0.f, 0.f, 0.f, 0.f, 0.f};

  for (int kc = 0; kc < D_MODEL; kc += KCH) {
    __syncthreads();
    for (int e = tid; e < MTILE * KCH; e += 256) {
      const int m = e >> 7, k = e & (KCH - 1);
      a_s[m * LDA + k] = comb[(size_t)(m0 + m) * D_MODEL + kc + k];
    }
    __syncthreads();
    for (int k0 = 0; k0 < KCH; k0 += 4) {
      const int kk = k0 + koff;
      const float* bp = W_in + (size_t)(kc + kk) * D_FF + nn;
      v2f b;
      b.x = rintf(bp[0] * s) * sinv;
      b.y = rintf(bp[D_FF] * s) * sinv;
      #pragma unroll
      for (int ms = 0; ms < 4; ++ms) {
        v2f a;
        a.x = a_s[(ms * 16 + l16) * LDA + kk];
        a.y = a_s[(ms * 16 + l16) * LDA + kk + 1];
        acc[ms] = __builtin_amdgcn_wmma_f32_16x16x4_f32(false, a, false, b, (short)0,
                                                        acc[ms], false, false);
      }
    }
  }
  #pragma unroll
  for (int ms = 0; ms < 4; ++ms) {
    float* outp = hidden + (size_t)(m0 + ms * 16 + half * 8) * D_FF + nn;
    #pragma unroll
    for (int r = 0; r < 8; ++r) outp[(size_t)r * D_FF] = acc[ms][r];
  }
}

// per-row mean/rstd of quantize_a16(hidden); wave per row
__global__ __launch_bounds__(256) void row_stats_kernel(const float* __restrict__ hidden,
                                                        const float* __restrict__ slots,
                                                        float* __restrict__ meanp,
                                                        float* __restrict__ rstdp) {
  const int row = blockIdx.x * 8 + (threadIdx.x >> 5);
  const int lane = threadIdx.x & 31;
  const float sh = 32767.0f / fmaxf(slots[4], 1e-8f), shi = 1.0f / sh;
  const float* hp = hidden + (size_t)row * D_FF;
  float s1 = 0.f, s2 = 0.f;
  for (int k = lane; k < D_FF; k += 32) {
    float hs = rintf(hp[k] * sh) * shi;
    s1 += hs; s2 += hs * hs;
  }
  s1 = wave_sum(s1); s2 = wave_sum(s2);
  if (lane == 0) {
    const float m = s1 * (1.0f / D_FF);
    const float var = s2 * (1.0f / D_FF) - m * m;
    meanp[row] = m;
    rstdp[row] = rsqrtf(var + LN_EPS);
  }
}

// out = LN(q16(hidden)) @ dequant_row(W_out) + q16(x) @ dequant(W_skip)
// block tile 64(M) x 128(N); same 4-C-tiles-per-wave scheme, dual K loops
__global__ __launch_bounds__(256) void gemm_out_kernel(const float* __restrict__ hidden,
                                                       const float* __restrict__ meanp,
                                                       const float* __restrict__ rstdp,
                                                       const float* __restrict__ lnw,
                                                       const float* __restrict__ lnb,
                                                       const float* __restrict__ W_out,
                                                       const float* __restrict__ rowmax,
                                                       const float* __restrict__ x,
                                                       const float* __restrict__ slots,
                                                       const float* __restrict__ W_skip,
                                                       float* __restrict__ out) {
  constexpr int MTILE = 64;
  constexpr int KCH = 128;
  constexpr int LDA = KCH + 2;
  __shared__ float a_s[MTILE * LDA];
  __shared__ float bs_s[KCH];    // per-k dequant scale for W_out rows
  __shared__ float bsi_s[KCH];
  const int tid = threadIdx.x;
  const int lane = tid & 31, wid = tid >> 5;
  const int m0 = blockIdx.x * MTILE;
  const int n0 = blockIdx.y * 128 + wid * 16;
  const int half = lane >> 4, l16 = lane & 15;
  const int nn = n0 + l16;
  const int koff = half * 2;

  const float sh = 32767.0f / fmaxf(slots[4], 1e-8f), shi = 1.0f / sh;
  const float sx = 32767.0f / fmaxf(slots[0], 1e-8f), sxi = 1.0f / sx;
  const float ssk = 127.0f / fmaxf(slots[3], 1e-8f), sski = 1.0f / ssk;

  v8f acc[4];
  #pragma unroll
  for (int ms = 0; ms < 4; ++ms) acc[ms] = v8f{0.f, 0.f, 0.f, 0.f, 0.f, 0.f, 0.f, 0.f};

  // Part 1: LN(q16(hidden)) @ W_out_sim, K = 4096
  for (int kc = 0; kc < D_FF; kc += KCH) {
    __syncthreads();
    for (int e = tid; e < MTILE * KCH; e += 256) {
      const int m = e >> 7, k = e & (KCH - 1);
      const int kg = kc + k;
      const int mg = m0 + m;
      float hs = rintf(hidden[(size_t)mg * D_FF + kg] * sh) * shi;
      a_s[m * LDA + k] = (hs - meanp[mg]) * rstdp[mg] * lnw[kg] + lnb[kg];
    }
    if (tid < KCH) {
      float sc = 127.0f / fmaxf(rowmax[kc + tid], 1e-8f);
      bs_s[tid] = sc; bsi_s[tid] = 1.0f / sc;
    }
    __syncthreads();
    for (int k0 = 0; k0 < KCH; k0 += 4) {
      const int kk = k0 + koff;
      const float* bp = W_out + (size_t)(kc + kk) * D_MODEL + nn;
      v2f b;
      b.x = rintf(bp[0] * bs_s[kk]) * bsi_s[kk];
      b.y = rintf(bp[D_MODEL] * bs_s[kk + 1]) * bsi_s[kk + 1];
      #pragma unroll
      for (int ms = 0; ms < 4; ++ms) {
        v2f a;
        a.x = a_s[(ms * 16 + l16) * LDA + kk];
        a.y = a_s[(ms * 16 + l16) * LDA + kk + 1];
        acc[ms] = __builtin_amdgcn_wmma_f32_16x16x4_f32(false, a, false, b, (short)0,
                                                        acc[ms], false, false);
      }
    }
  }

  // Part 2: q16(x) @ W_skip_sim, K = 1024, same accumulators
  for (int kc = 0; kc < D_MODEL; kc += KCH) {
    __syncthreads();
    for (int e = tid; e < MTILE * KCH; e += 256) {
      const int m = e >> 7, k = e & (KCH - 1);
      a_s[m * LDA + k] = rintf(x[(size_t)(m0 + m) * D_MODEL + kc + k] * sx) * sxi;
    }
    __syncthreads();
    for (int k0 = 0; k0 < KCH; k0 += 4) {
      const int kk = k0 + koff;
      const float* bp = W_skip + (size_t)(kc + kk) * D_MODEL + nn;
      v2f b;
      b.x = rintf(bp[0] * ssk) * sski;
      b.y = rintf(bp[D_MODEL] * ssk) * sski;
      #pragma unroll
      for (int ms = 0; ms < 4; ++ms) {
        v2f a;
        a.x = a_s[(ms * 16 + l16) * LDA + kk];
        a.y = a_s[(ms * 16 + l16) * LDA + kk + 1];
        acc[ms] = __builtin_amdgcn_wmma_f32_16x16x4_f32(false, a, false, b, (short)0,
                                                        acc[ms], false, false);
      }
    }
  }

  #pragma unroll
  for (int ms = 0; ms < 4; ++ms) {
    float* outp = out + (size_t)(m0 + ms * 16 + half * 8) * D_MODEL + nn;
    #pragma unroll
    for (int r = 0; r < 8; ++r) outp[(size_t)r * D_MODEL] = acc[ms][r];
  }
}

extern "C" void kernel_launch(void* const* d_in, const int* in_sizes, int n_in,
                              void* d_out, int out_size, void* d_ws, size_t ws_size,
                              hipStream_t stream) {
  (void)in_sizes; (void)n_in; (void)out_size; (void)ws_size;
  const float* x       = (const float*)d_in[0];
  const float* W_K     = (const float*)d_in[1];
  const float* W_V     = (const float*)d_in[2];
  const float* W_in    = (const float*)d_in[3];
  const float* W_out   = (const float*)d_in[4];
  const float* W_skip  = (const float*)d_in[5];
  const float* ln_in_w = (const float*)d_in[6];
  const float* ln_in_b = (const float*)d_in[7];
  const float* ln_out_w= (const float*)d_in[8];
  const float* ln_out_b= (const float*)d_in[9];
  const int*   ids     = (const int*)d_in[10];
  float* out = (float*)d_out;
  float* ws  = (float*)d_ws;

  float* slots  = ws + OFF_SLOTS;
  float* colmax = ws + OFF_COLMAX;
  float* rowmax = ws + OFF_ROWMAX;
  float* meanp  = ws + OFF_MEAN;
  float* rstdp  = ws + OFF_RSTD;
  float* A      = ws + OFF_A;
  float* comb   = ws + OFF_COMB;
  float* hidden = ws + OFF_HIDDEN;

  init_slots_kernel<<<1, 64, 0, stream>>>(slots);
  amax_kernel<<<1024, 256, 0, stream>>>(x, (long)NTOK * D_MODEL, slots + 0);
  amax_kernel<<<4096, 256, 0, stream>>>(W_K, (long)D_MODEL * N_EMB, slots + 1);
  amax_kernel<<<4096, 256, 0, stream>>>(W_V, (long)D_MODEL * N_EMB, slots + 2);
  amax_kernel<<<512, 256, 0, stream>>>(W_skip, (long)D_MODEL * D_MODEL, slots + 3);
  colmax_kernel<<<D_FF / 256, 256, 0, stream>>>(W_in, colmax);
  rowmax_kernel<<<D_FF / 8, 256, 0, stream>>>(W_out, rowmax);

  ln_in_kernel<<<NTOK, 256, 0, stream>>>(x, slots, ln_in_w, ln_in_b, A);
  attn_kernel<<<NTOK, 256, 0, stream>>>(A, W_K, W_V, ids, slots, comb);

  gemm_hidden_kernel<<<dim3(NTOK / 64, D_FF / 128), 256, 0, stream>>>(comb, W_in, colmax, hidden);

  amax_kernel<<<2048, 256, 0, stream>>>(hidden, (long)NTOK * D_FF, slots + 4);
  row_stats_kernel<<<NTOK / 8, 256, 0, stream>>>(hidden, slots, meanp, rstdp);

  gemm_out_kernel<<<dim3(NTOK / 64, D_MODEL / 128), 256, 0, stream>>>(
      hidden, meanp, rstdp, ln_out_w, ln_out_b, W_out, rowmax, x, slots, W_skip, out);
}